// MaskedConv2d_58437325030077
// MI455X (gfx1250) — compile-verified
//
#include <hip/hip_runtime.h>
#include <hip/hip_bf16.h>

typedef __attribute__((ext_vector_type(2))) float v2f;
typedef __attribute__((ext_vector_type(4))) float v4f;
typedef __attribute__((ext_vector_type(8))) float v8f;

#define KSEL 1152      // selected K rows (C*9/2)
#define OC   512       // output channels
#define CIN  256
#define HIMG 32
#define WIMG 32
#define OWID 30
#define PP   900       // 30*30 output pixels per image
#define BM   128
#define BN   256
#define BK   16
#define ASTR 20        // padded LDS stride for A tile (floats)
#define BSTR 264       // padded LDS stride for B tile (floats); 2*264 % 64 == 16

__global__ __launch_bounds__(256)
void masked_conv_wmma_f32_v2(const float* __restrict__ x,
                             const float* __restrict__ w,
                             const int*   __restrict__ idx,
                             float* __restrict__ out,
                             int M_tot)
{
    __shared__ float Alds[BM * ASTR];   // 128 x 16 (padded)
    __shared__ float Blds[BK * BSTR];   // 16 x 256 (padded)
    __shared__ int   koff[KSEL];        // gather offsets per selected k
    __shared__ int   mbase[BM];         // per-row global base into x (-1 = OOB)

    const int tid = threadIdx.x;
    const int m0  = blockIdx.x * BM;
    const int n0  = blockIdx.y * BN;

    // One-time: decode idx -> element offset within an image.
    for (int k = tid; k < KSEL; k += 256) {
        int r   = idx[k];
        int c   = r / 9;
        int rem = r - c * 9;
        int ki  = rem / 3;
        int kj  = rem - ki * 3;
        koff[k] = c * (HIMG * WIMG) + ki * WIMG + kj;
    }
    // One-time: per-row base address (im2col column -> x base).
    if (tid < BM) {
        int m = m0 + tid;
        int v = -1;
        if (m < M_tot) {
            int nimg = m / PP;
            int p    = m - nimg * PP;
            int oy   = p / OWID;
            int ox   = p - oy * OWID;
            v = nimg * (CIN * HIMG * WIMG) + oy * WIMG + ox;
        }
        mbase[tid] = v;
    }
    __syncthreads();

    // Register-cache loader assignments (constant across K stages).
    const int my_kk   = tid & 15;       // A: k within tile
    const int my_rowb = tid >> 4;       // A: row = e*16 + my_rowb
    int my_mb[8];
    #pragma unroll
    for (int e = 0; e < 8; ++e) my_mb[e] = mbase[e * 16 + my_rowb];
    const int my_c4 = (tid & 63) * 4;   // B: float4 column within tile
    const int my_kr = tid >> 6;         // B: k row = e*4 + my_kr
    const bool full = (m0 + BM) <= M_tot;   // uniform: block has no OOB rows

    const int lane = tid & 31;
    const int half = lane >> 4;         // half-wave select
    const int lrow = lane & 15;
    const int wave = tid >> 5;
    const int wm   = wave >> 2;         // 0..1 : wave M position
    const int wn   = wave & 3;          // 0..3 : wave N position

    v8f acc[4][4];
    #pragma unroll
    for (int i = 0; i < 4; ++i)
        #pragma unroll
        for (int j = 0; j < 4; ++j)
            acc[i][j] = (v8f)(0.0f);

    for (int k0 = 0; k0 < KSEL; k0 += BK) {
        __syncthreads();   // previous stage's compute done

        // Stage A tile (gathered loads; uniform branch skips OOB guard).
        const int ko = koff[k0 + my_kk];
        if (full) {
            #pragma unroll
            for (int e = 0; e < 8; ++e)
                Alds[(e * 16 + my_rowb) * ASTR + my_kk] = x[my_mb[e] + ko];
        } else {
            #pragma unroll
            for (int e = 0; e < 8; ++e) {
                const int mb = my_mb[e];
                float v = 0.0f;
                if (mb >= 0) v = x[mb + ko];
                Alds[(e * 16 + my_rowb) * ASTR + my_kk] = v;
            }
        }
        // Stage B tile as float4 (b128 load + b128 LDS store) + prefetch next.
        #pragma unroll
        for (int e = 0; e < 4; ++e) {
            const int kr = e * 4 + my_kr;
            *(v4f*)&Blds[kr * BSTR + my_c4] =
                *(const v4f*)&w[(k0 + kr) * OC + n0 + my_c4];
        }
        if (k0 + BK < KSEL)
            __builtin_prefetch(&w[(k0 + BK + my_kr) * OC + n0 + my_c4], 0, 0);
        __syncthreads();

        // 4 k-steps x (4 M-frags x 4 N-frags) of V_WMMA_F32_16X16X4_F32.
        #pragma unroll
        for (int kk4 = 0; kk4 < 4; ++kk4) {
            const int kb = kk4 * 4 + half * 2;   // half-wave K pair select
            v2f bfrag[4];
            #pragma unroll
            for (int j = 0; j < 4; ++j) {
                const int n = wn * 64 + j * 16 + lrow;
                bfrag[j].x = Blds[kb * BSTR + n];         // K even row
                bfrag[j].y = Blds[(kb + 1) * BSTR + n];   // K odd row
            }
            #pragma unroll
            for (int i = 0; i < 4; ++i) {
                const v2f a =
                    *(const v2f*)&Alds[(wm * 64 + i * 16 + lrow) * ASTR + kb];
                #pragma unroll
                for (int j = 0; j < 4; ++j) {
                    acc[i][j] = __builtin_amdgcn_wmma_f32_16x16x4_f32(
                        /*neg_a=*/false, a, /*neg_b=*/false, bfrag[j],
                        /*c_mod=*/(short)0, acc[i][j],
                        /*reuse_a=*/false, /*reuse_b=*/false);
                }
            }
        }
    }

    // Write-out with NCHW transpose: out[(nimg*OC + oc)*900 + p].
    #pragma unroll
    for (int i = 0; i < 4; ++i) {
        #pragma unroll
        for (int j = 0; j < 4; ++j) {
            const int oc = n0 + wn * 64 + j * 16 + lrow;
            #pragma unroll
            for (int v = 0; v < 8; ++v) {
                const int m = m0 + wm * 64 + i * 16 + half * 8 + v;
                if (m < M_tot) {
                    const int nimg = m / PP;
                    const int p    = m - nimg * PP;
                    out[(nimg * OC + oc) * PP + p] = acc[i][j][v];
                }
            }
        }
    }
}

extern "C" void kernel_launch(void* const* d_in, const int* in_sizes, int n_in,
                              void* d_out, int out_size, void* d_ws, size_t ws_size,
                              hipStream_t stream) {
    const float* x   = (const float*)d_in[0];
    const float* w   = (const float*)d_in[1];
    const int*   idx = (const int*)d_in[2];
    float*       out = (float*)d_out;

    const int n_img = in_sizes[0] / (CIN * HIMG * WIMG);   // 16
    const int M_tot = n_img * PP;                          // 14400

    dim3 grid((M_tot + BM - 1) / BM, OC / BN);             // 113 x 2
    masked_conv_wmma_f32_v2<<<grid, 256, 0, stream>>>(x, w, idx, out, M_tot);
}